// VariationalGraphSageEncoder_17952963297838
// MI455X (gfx1250) — compile-verified
//
#include <hip/hip_runtime.h>
#include <hip/hip_bf16.h>

typedef float v2f __attribute__((ext_vector_type(2)));
typedef float v8f __attribute__((ext_vector_type(8)));

#define NNODES 100000
#define NEDGES 1600000
#define CNT_PAD 100352   // N rounded up to 256B multiple (in floats)

// ---------------------------------------------------------------------------
// Edge scatter: one wave per edge. Each lane moves 4 channels (float4 gather,
// 4 f32 atomic adds). Gathers/atomics resolve in L2 (x/h are 51.2MB << 192MB).
// Lane 0 optionally accumulates the in-degree as float.
// ---------------------------------------------------------------------------
__global__ __launch_bounds__(256) void sage_scatter(
    const float* __restrict__ X, const int* __restrict__ src,
    const int* __restrict__ dst, float* __restrict__ agg,
    float* __restrict__ cnt, int E)
{
    int wave = (int)((blockIdx.x * blockDim.x + threadIdx.x) >> 5);
    int lane = (int)(threadIdx.x & 31);
    if (wave >= E) return;
    int s = src[wave];
    int d = dst[wave];
    const float4 v = *(const float4*)(X + s * 128 + lane * 4);
    float* a = agg + d * 128 + lane * 4;
    atomicAdd(a + 0, v.x);
    atomicAdd(a + 1, v.y);
    atomicAdd(a + 2, v.z);
    atomicAdd(a + 3, v.w);
    if (cnt != nullptr && lane == 0) atomicAdd(cnt + d, 1.0f);
}

// cnt[i] <- 1 / max(cnt[i], 1)   (division hoisted out of the GEMM kernels)
__global__ __launch_bounds__(256) void invert_cnt(float* __restrict__ cnt, int n)
{
    int i = (int)(blockIdx.x * blockDim.x + threadIdx.x);
    if (i < n) cnt[i] = 1.0f / fmaxf(cnt[i], 1.0f);
}

// Wt[n*K + k] = W[k*N + n]  (K=128 rows in, N cols in) — makes B fragments
// {K,K+1} contiguous so the GEMM can use b64 loads with immediate offsets.
__global__ __launch_bounds__(256) void transpose_w(
    const float* __restrict__ W, float* __restrict__ Wt, int N /*cols*/)
{
    int i = (int)(blockIdx.x * blockDim.x + threadIdx.x);
    if (i >= 128 * N) return;
    int k = i / N;
    int n = i - k * N;
    Wt[n * 128 + k] = W[i];
}

// ---------------------------------------------------------------------------
// Fused SAGE GEMM tile kernel (wave32 WMMA f32 16x16x4):
//   out = act( (Aagg * invcnt) @ Wl  +  Ax @ Wr  +  bias )
// One wave = one 16x16 output tile; K=128 on both paths accumulated into one
// v8f accumulator (64 WMMAs/tile). NCOLS is compile-time so all address math
// folds to base-pointer + immediate-offset b64 loads.
//
// Fragment layouts (ISA 7.12.2, wave32):
//   A 16x4 f32 : lane<16 -> row=lane,    regs {K, K+1}; lanes>=16 -> {K+2,K+3}
//   B 4x16 f32 : lane<16 -> col=lane,    regs {K, K+1}; lanes>=16 -> {K+2,K+3}
//   C/D 16x16  : col = lane&15; reg r -> row = r (+8 for lanes 16..31)
// WlT/WrT are transposed weights (NCOLS x 128 row-major).
// ---------------------------------------------------------------------------
template<int NCOLS, int RELU>
__global__ __launch_bounds__(256) void sage_gemm(
    const float* __restrict__ Aagg, const float* __restrict__ invcnt,
    const float* __restrict__ Ax,
    const float* __restrict__ WlT, const float* __restrict__ WrT,
    const float* __restrict__ bias, float* __restrict__ out, int nrows)
{
    int wave = (int)((blockIdx.x * blockDim.x + threadIdx.x) >> 5);
    int lane = (int)(threadIdx.x & 31);
    const int tiles_n = NCOLS >> 4;
    int tm = wave / tiles_n;
    int tn = wave - tm * tiles_n;
    if (tm >= (nrows >> 4)) return;  // whole-wave uniform exit: EXEC stays all-1s

    int hi  = (lane >> 4);           // 0 for lanes 0..15, 1 for lanes 16..31
    int khi = hi << 1;               // K offset within the 4-wide K slab
    int row = tm * 16 + (lane & 15);
    int col = tn * 16 + (lane & 15);

    float invc = invcnt[row];
    // per-lane base pointers; all inner-loop offsets are immediates
    const float* arow = Aagg + row * 128 + khi;
    const float* xrow = Ax   + row * 128 + khi;
    const float* wl   = WlT  + col * 128 + khi;
    const float* wr   = WrT  + col * 128 + khi;

    v8f c = {};
    // aggregated-neighbor path: (agg * invc) @ Wl
    #pragma unroll
    for (int k0 = 0; k0 < 128; k0 += 4) {
        v2f a = *(const v2f*)(arow + k0);
        v2f b = *(const v2f*)(wl + k0);
        a.x *= invc; a.y *= invc;
        c = __builtin_amdgcn_wmma_f32_16x16x4_f32(false, a, false, b,
                                                  (short)0, c, false, false);
    }
    // root path: x @ Wr
    #pragma unroll
    for (int k0 = 0; k0 < 128; k0 += 4) {
        v2f a = *(const v2f*)(xrow + k0);
        v2f b = *(const v2f*)(wr + k0);
        c = __builtin_amdgcn_wmma_f32_16x16x4_f32(false, a, false, b,
                                                  (short)0, c, false, false);
    }

    float bv = bias[col];
    int obase = (tm * 16 + (hi << 3)) * NCOLS + col;
    #pragma unroll
    for (int r = 0; r < 8; ++r) {
        float v = c[r] + bv;
        if (RELU) v = fmaxf(v, 0.0f);
        out[obase + r * NCOLS] = v;
    }
}

// ---------------------------------------------------------------------------
// Orchestration (graph-capture safe: kernel launches + hipMemsetAsync only).
// Workspace (floats): [cnt: CNT_PAD][agg: N*128][h: N*128][6 transposed W]
// ---------------------------------------------------------------------------
extern "C" void kernel_launch(void* const* d_in, const int* in_sizes, int n_in,
                              void* d_out, int out_size, void* d_ws, size_t ws_size,
                              hipStream_t stream) {
    (void)in_sizes; (void)n_in; (void)out_size; (void)ws_size;

    const float* x   = (const float*)d_in[0];
    const int*   ei  = (const int*)d_in[1];      // (2, E): row0 = src, row1 = dst
    const float* W1l = (const float*)d_in[2];
    const float* b1  = (const float*)d_in[3];
    const float* W1r = (const float*)d_in[4];
    const float* Wml = (const float*)d_in[5];
    const float* bm  = (const float*)d_in[6];
    const float* Wmr = (const float*)d_in[7];
    const float* Wsl = (const float*)d_in[8];
    const float* bs  = (const float*)d_in[9];
    const float* Wsr = (const float*)d_in[10];
    float* out = (float*)d_out;

    const int N = NNODES, E = NEDGES;
    const int* src = ei;
    const int* dst = ei + E;

    float* ws   = (float*)d_ws;
    float* cnt  = ws;                          // N (padded to CNT_PAD)
    float* agg  = ws + CNT_PAD;                // N*128, reused for both layers
    float* h    = agg + (size_t)N * 128;       // N*128
    float* wT   = h   + (size_t)N * 128;       // transposed weights
    float* W1lT = wT;                          // 128*128
    float* W1rT = W1lT + 128 * 128;
    float* WmlT = W1rT + 128 * 128;            // 64*128 each from here
    float* WmrT = WmlT + 64 * 128;
    float* WslT = WmrT + 64 * 128;
    float* WsrT = WslT + 64 * 128;

    dim3 blk(256);

    // zero cnt + agg in one contiguous memset
    hipMemsetAsync(ws, 0, (CNT_PAD + (size_t)N * 128) * sizeof(float), stream);

    // transpose the six weight matrices (tiny)
    transpose_w<<<(128 * 128 + 255) / 256, blk, 0, stream>>>(W1l, W1lT, 128);
    transpose_w<<<(128 * 128 + 255) / 256, blk, 0, stream>>>(W1r, W1rT, 128);
    transpose_w<<<(128 * 64 + 255) / 256, blk, 0, stream>>>(Wml, WmlT, 64);
    transpose_w<<<(128 * 64 + 255) / 256, blk, 0, stream>>>(Wmr, WmrT, 64);
    transpose_w<<<(128 * 64 + 255) / 256, blk, 0, stream>>>(Wsl, WslT, 64);
    transpose_w<<<(128 * 64 + 255) / 256, blk, 0, stream>>>(Wsr, WsrT, 64);

    int scat_blocks = (E + 7) / 8;             // 8 waves (edges) per block

    // layer 1: aggregate x, compute degrees, then invert degrees in place
    sage_scatter<<<scat_blocks, blk, 0, stream>>>(x, src, dst, agg, cnt, E);
    invert_cnt<<<(N + 255) / 256, blk, 0, stream>>>(cnt, N);

    // h = relu(mean(agg) @ W1l + b1 + x @ W1r)   (N x 128)
    {
        int waves = (N / 16) * (128 / 16);     // 50000
        sage_gemm<128, 1><<<(waves + 7) / 8, blk, 0, stream>>>(
            agg, cnt, x, W1lT, W1rT, b1, h, N);
    }

    // re-zero agg, aggregate h (degrees unchanged — same graph)
    hipMemsetAsync(agg, 0, (size_t)N * 128 * sizeof(float), stream);
    sage_scatter<<<scat_blocks, blk, 0, stream>>>(h, src, dst, agg, nullptr, E);

    // mu and logstd share the aggregated h
    {
        int waves = (N / 16) * (64 / 16);      // 25000
        sage_gemm<64, 0><<<(waves + 7) / 8, blk, 0, stream>>>(
            agg, cnt, h, WmlT, WmrT, bm, out, N);
        sage_gemm<64, 0><<<(waves + 7) / 8, blk, 0, stream>>>(
            agg, cnt, h, WslT, WsrT, bs, out + (size_t)N * 64, N);
    }
}